// Loss_40080634806416
// MI455X (gfx1250) — compile-verified
//
#include <hip/hip_runtime.h>
#include <math.h>

// ---------------------------------------------------------------------------
// Reference is a pure data-movement op:
//   out[:,0:4] = broadcast scalars (from x[0,0..1], y[0,0..1])
//   out[:,4]   = 0
//   out[:,5:]  = y[:,5:]
// Memory-bound: ~340 MB @ 23.3 TB/s => ~15 us floor. No matrix math -> no
// WMMA. Use CDNA5 async global<->LDS copy engine (ASYNCcnt) for the bulk
// stream; patch the 5 head columns in LDS between load and store.
// ---------------------------------------------------------------------------

#define D_COLS        85
#define THREADS       256
#define ROWS_PER_TILE 64
#define ROW_BYTES     (D_COLS * 4)                 // 340 B
#define TILE_BYTES    (ROWS_PER_TILE * ROW_BYTES)  // 21760 B (16B-multiple)
#define CHUNK         16                           // b128 per lane
#define STRIDE        (THREADS * CHUNK)            // 4096 B per block step

#if defined(__gfx1250__) &&                                                   \
    __has_builtin(__builtin_amdgcn_global_load_async_to_lds_b128) &&          \
    __has_builtin(__builtin_amdgcn_global_store_async_from_lds_b128)
#define USE_ASYNC_BUILTINS 1
#endif

// Builtin signature (from hipcc diagnostic): first param is
// 'int __attribute__((vector_size(16))) *' in the global address space.
typedef int v4i __attribute__((vector_size(4 * sizeof(int))));
typedef __attribute__((address_space(1))) v4i as1_v4i;
typedef __attribute__((address_space(3))) v4i as3_v4i;
typedef __attribute__((address_space(3))) char as3_char;

__device__ __forceinline__ void async_g2l_16(const char* g, char* l) {
#if defined(USE_ASYNC_BUILTINS)
  __builtin_amdgcn_global_load_async_to_lds_b128(
      (as1_v4i*)g, (as3_v4i*)l, 0, 0);
#else
  unsigned lofs = (unsigned)(unsigned long long)(as3_char*)l;
  asm volatile("global_load_async_to_lds_b128 %0, %1, off"
               :: "v"(lofs), "v"(g) : "memory");
#endif
}

__device__ __forceinline__ void async_l2g_16(char* g, const char* l) {
#if defined(USE_ASYNC_BUILTINS)
  __builtin_amdgcn_global_store_async_from_lds_b128(
      (as1_v4i*)g, (as3_v4i*)l, 0, 0);
#else
  unsigned lofs = (unsigned)(unsigned long long)(as3_char*)l;
  asm volatile("global_store_async_from_lds_b128 %0, %1, off"
               :: "v"(g), "v"(lofs) : "memory");
#endif
}

__device__ __forceinline__ void wait_async_zero() {
#if defined(__gfx1250__) && __has_builtin(__builtin_amdgcn_s_wait_asynccnt)
  __builtin_amdgcn_s_wait_asynccnt(0);
#else
  asm volatile("s_wait_asynccnt 0" ::: "memory");
#endif
}

__global__ __launch_bounds__(THREADS)
void loss_head_copy_kernel(const float* __restrict__ x,
                           const float* __restrict__ y,
                           float* __restrict__ out,
                           int n_rows) {
  __shared__ float tile[TILE_BYTES / 4];  // 5440 floats = 21.25 KB
  __shared__ float head[8];

  const int       tid  = (int)threadIdx.x;
  const long long row0 = (long long)blockIdx.x * ROWS_PER_TILE;
  if (row0 >= n_rows) return;  // block-uniform

  const int rows      = (n_rows - row0) < ROWS_PER_TILE
                            ? (int)(n_rows - row0) : ROWS_PER_TILE;
  const int tileBytes = rows * ROW_BYTES;

  // Head scalars: bx,by use sigmoid; bw,bh replicate the source's
  // AC*exp(px)/exp(py) (its apparent bug is part of the spec).
  if (tid == 0) {
    const float px = x[0], py = x[1];
    const float tx = y[0], ty = y[1];
    head[0] = 1.0f / (1.0f + expf(-px)) + tx;   // bx
    head[1] = 1.0f / (1.0f + expf(-py)) + ty;   // by
    head[2] = 1e-16f * expf(px);                // bw
    head[3] = 1e-16f * expf(py);                // bh
    head[4] = 0.0f;                             // objectness column
  }

  const char* gsrc  = (const char*)y   + row0 * ROW_BYTES;
  char*       gdst  = (char*)out       + row0 * ROW_BYTES;
  char*       lbase = (char*)tile;

  // Stage the whole tile HBM -> LDS with async B128 transfers (ASYNCcnt).
  for (int off = tid * CHUNK; off < tileBytes; off += STRIDE)
    async_g2l_16(gsrc + off, lbase + off);
  wait_async_zero();
  __syncthreads();  // all waves' loads visible in LDS; head[] visible too

  // Patch columns 0..4 of every row in LDS.
  for (int i = tid; i < rows * 5; i += THREADS) {
    const int r = i / 5;
    const int c = i - 5 * r;
    tile[r * D_COLS + c] = head[c];
  }
  __syncthreads();  // patch visible before streaming out

  // Stream the patched tile LDS -> HBM with async B128 transfers.
  for (int off = tid * CHUNK; off < tileBytes; off += STRIDE)
    async_l2g_16(gdst + off, lbase + off);
  wait_async_zero();  // tile's stores retired before block exits
}

extern "C" void kernel_launch(void* const* d_in, const int* in_sizes, int n_in,
                              void* d_out, int out_size, void* d_ws, size_t ws_size,
                              hipStream_t stream) {
  (void)n_in; (void)d_ws; (void)ws_size; (void)in_sizes;
  const float* x   = (const float*)d_in[0];
  const float* y   = (const float*)d_in[1];
  float*       out = (float*)d_out;

  const int n_rows = out_size / D_COLS;  // 500000
  const int tiles  = (n_rows + ROWS_PER_TILE - 1) / ROWS_PER_TILE;

  loss_head_copy_kernel<<<dim3(tiles), dim3(THREADS), 0, stream>>>(
      x, y, out, n_rows);
}